// Note_decoder_33191507263663
// MI455X (gfx1250) — compile-verified
//
#include <hip/hip_runtime.h>
#include <math.h>

#define H 256          // hidden size
#define NB 512         // attention partial blocks
#define IN 50          // LSTM input width

typedef __attribute__((ext_vector_type(2))) float v2f;
typedef __attribute__((ext_vector_type(8))) float v8f;

// ---------------------------------------------------------------------------
// Kernel 1: flash-style fused scores+softmax+context partials.
// Reads encoder_outputs (134 MB) exactly once -> HBM-roofline bound (~6 us).
// Each wave handles 32 contiguous rows; row loads are coalesced float4s.
// Per-block output: (running max m, running sum s, rescaled ctx[256]).
// ---------------------------------------------------------------------------
__global__ __launch_bounds__(256) void attn_partial_kernel(
    const float* __restrict__ E, const float* __restrict__ hn,
    float* __restrict__ ws_m, float* __restrict__ ws_s,
    float* __restrict__ ws_acc, int L) {
  const int lane = threadIdx.x & 31;
  const int wave = threadIdx.x >> 5;

  // lane c owns columns [4c,4c+4) and [128+4c,128+4c+4)
  const float4 h0 = *reinterpret_cast<const float4*>(hn + lane * 4);
  const float4 h1 = *reinterpret_cast<const float4*>(hn + 128 + lane * 4);

  float4 a0 = make_float4(0.f, 0.f, 0.f, 0.f);
  float4 a1 = make_float4(0.f, 0.f, 0.f, 0.f);
  float m = -INFINITY, s = 0.f;

  const int rowsPerBlock = (L + NB - 1) / NB;
  const int base = blockIdx.x * rowsPerBlock;
  const int lim = min(base + rowsPerBlock, L);
  const int rowsPerWave = (rowsPerBlock + 7) / 8;
  const int r0 = base + wave * rowsPerWave;
  const int r1 = min(r0 + rowsPerWave, lim);

  for (int r = r0; r < r1; ++r) {
    const float* row = E + (size_t)r * H;
    if (r + 4 < r1) __builtin_prefetch(row + 4 * H + lane * 4, 0, 3);
    const float4 e0 = *reinterpret_cast<const float4*>(row + lane * 4);
    const float4 e1 = *reinterpret_cast<const float4*>(row + 128 + lane * 4);
    float part = e0.x * h0.x + e0.y * h0.y + e0.z * h0.z + e0.w * h0.w +
                 e1.x * h1.x + e1.y * h1.y + e1.z * h1.z + e1.w * h1.w;
    // wave32 all-reduce of the dot product
    #pragma unroll
    for (int off = 16; off > 0; off >>= 1) part += __shfl_xor(part, off, 32);

    const float nm = fmaxf(m, part);
    const float scale = __expf(m - nm);   // exp(-inf)=0 on first iter: safe
    const float p = __expf(part - nm);
    s = s * scale + p;
    a0.x = a0.x * scale + p * e0.x; a0.y = a0.y * scale + p * e0.y;
    a0.z = a0.z * scale + p * e0.z; a0.w = a0.w * scale + p * e0.w;
    a1.x = a1.x * scale + p * e1.x; a1.y = a1.y * scale + p * e1.y;
    a1.z = a1.z * scale + p * e1.z; a1.w = a1.w * scale + p * e1.w;
    m = nm;
  }

  // combine 8 waves within the block
  __shared__ float l_acc[8][H];
  __shared__ float l_m[8];
  __shared__ float l_s[8];
  *reinterpret_cast<float4*>(&l_acc[wave][lane * 4]) = a0;
  *reinterpret_cast<float4*>(&l_acc[wave][128 + lane * 4]) = a1;
  if (lane == 0) { l_m[wave] = m; l_s[wave] = s; }
  __syncthreads();

  float M = l_m[0];
  #pragma unroll
  for (int w = 1; w < 8; ++w) M = fmaxf(M, l_m[w]);
  const int j = threadIdx.x;  // one column per thread
  float col = 0.f;
  #pragma unroll
  for (int w = 0; w < 8; ++w) col += __expf(l_m[w] - M) * l_acc[w][j];
  ws_acc[(size_t)blockIdx.x * H + j] = col;
  if (threadIdx.x == 0) {
    float S = 0.f;
    #pragma unroll
    for (int w = 0; w < 8; ++w) S += __expf(l_m[w] - M) * l_s[w];
    ws_m[blockIdx.x] = M;
    ws_s[blockIdx.x] = S;
  }
}

__device__ __forceinline__ float sigmoidf_(float x) {
  return 1.f / (1.f + __expf(-x));
}

// ---------------------------------------------------------------------------
// Kernel 2 (1 block, 256 threads): cross-block softmax reduce -> ctx,
// attention projection, embedding gathers, LSTM cell, dual heads.
// W_hh @ h (1024x256 matvec) runs on V_WMMA_F32_16X16X4_F32:
//   64 tiles of 16 gates; B = h chunk broadcast across all 16 N columns so
//   every D column holds the gate value. Layouts per CDNA5 ISA 7.12.2:
//   A(16x4 f32): lanes 0-15 -> {K=k,k+1}, lanes 16-31 -> {K=k+2,k+3}, M=lane&15
//   D: VGPR v -> M=v (lanes 0-15) / M=v+8 (lanes 16-31), N=lane%16.
// ---------------------------------------------------------------------------
__global__ __launch_bounds__(256) void decoder_tail_kernel(
    const float* __restrict__ ws_m, const float* __restrict__ ws_s,
    const float* __restrict__ ws_acc,
    const int* __restrict__ sec_p, const int* __restrict__ chord_p,
    const int* __restrict__ true_pitch_p, const int* __restrict__ note,
    const float* __restrict__ hn, const float* __restrict__ cn,
    const float* __restrict__ chord_emb, const float* __restrict__ pitch_emb,
    const float* __restrict__ dur_emb,
    const float* __restrict__ W_ih, const float* __restrict__ W_hh,
    const float* __restrict__ b_ih, const float* __restrict__ b_hh,
    const float* __restrict__ pitch_W, const float* __restrict__ pitch_b,
    const float* __restrict__ dur_W, const float* __restrict__ dur_b,
    const float* __restrict__ attn_W, const float* __restrict__ attn_b,
    float* __restrict__ out) {
  const int t = threadIdx.x;
  const int lane = t & 31;
  const int wave = t >> 5;

  __shared__ float lds_h[H];
  __shared__ float lds_ctx[H];
  __shared__ float lds_scale[NB];
  __shared__ float lds_x[IN];
  __shared__ float lds_gates[4 * H];
  __shared__ float lds_hnew[H];
  __shared__ float red[256];
  __shared__ float sg_sh;

  lds_h[t] = hn[t];

  // ---- global softmax max over NB block partials ----
  float lm = fmaxf(ws_m[t], ws_m[t + 256]);
  red[t] = lm;
  __syncthreads();
  #pragma unroll
  for (int st = 128; st > 0; st >>= 1) {
    if (t < st) red[t] = fmaxf(red[t], red[t + st]);
    __syncthreads();
  }
  const float Mg = red[0];
  __syncthreads();

  lds_scale[t] = __expf(ws_m[t] - Mg);
  lds_scale[t + 256] = __expf(ws_m[t + 256] - Mg);
  red[t] = lds_scale[t] * ws_s[t] + lds_scale[t + 256] * ws_s[t + 256];
  __syncthreads();
  #pragma unroll
  for (int st = 128; st > 0; st >>= 1) {
    if (t < st) red[t] += red[t + st];
    __syncthreads();
  }
  if (t == 0) sg_sh = red[0];
  __syncthreads();
  const float inv_sg = 1.f / sg_sh;

  // ---- ctx[j] = sum_b scale_b * acc_b[j] / s_g  (coalesced over threads) ----
  float cj = 0.f;
  for (int b = 0; b < NB; ++b) cj += lds_scale[b] * ws_acc[(size_t)b * H + t];
  lds_ctx[t] = cj * inv_sg;
  __syncthreads();

  // ---- build x[50] = [e_pitch, e_dur, e_chord, e_sec, ctx@attn_W.T+b] ----
  const int sec = sec_p[0], chord = chord_p[0], tp = true_pitch_p[0];
  const int n0 = note[0], n1 = note[1];
  if (t < 10) {
    float acc = attn_b[t];
    for (int c = 0; c < H; ++c) acc += attn_W[t * H + c] * lds_ctx[c];
    lds_x[40 + t] = acc;
  } else if (t < 20) {
    lds_x[t - 10] = pitch_emb[n0 * 10 + (t - 10)];
  } else if (t < 30) {
    lds_x[10 + (t - 20)] = dur_emb[n1 * 10 + (t - 20)];
  } else if (t < 40) {
    lds_x[20 + (t - 30)] = chord_emb[chord * 10 + (t - 30)];
  } else if (t < 50) {
    lds_x[30 + (t - 40)] = chord_emb[sec * 10 + (t - 40)];
  }
  __syncthreads();

  // ---- W_hh @ h via f32 WMMA: wave w -> tiles w*8 .. w*8+7 ----
  for (int tt = 0; tt < 8; ++tt) {
    const int tile = wave * 8 + tt;
    const int rowA = tile * 16 + (lane & 15);
    const int khalf = (lane >> 4) << 1;  // 0 for lanes 0-15, 2 for 16-31
    v8f acc;
    acc[0]=0.f; acc[1]=0.f; acc[2]=0.f; acc[3]=0.f;
    acc[4]=0.f; acc[5]=0.f; acc[6]=0.f; acc[7]=0.f;
    #pragma unroll 4
    for (int k = 0; k < H; k += 4) {
      const int kk = k + khalf;
      v2f a = *reinterpret_cast<const v2f*>(W_hh + (size_t)rowA * H + kk);
      v2f b;
      b.x = lds_h[kk];
      b.y = lds_h[kk + 1];
      acc = __builtin_amdgcn_wmma_f32_16x16x4_f32(
          false, a, false, b, (short)0, acc, false, false);
    }
    // every N column of D holds the gate value; lanes 0 and 16 write M=0..7/8..15
    if (lane == 0) {
      #pragma unroll
      for (int v = 0; v < 8; ++v) lds_gates[tile * 16 + v] = acc[v];
    } else if (lane == 16) {
      #pragma unroll
      for (int v = 0; v < 8; ++v) lds_gates[tile * 16 + 8 + v] = acc[v];
    }
  }
  __syncthreads();

  // ---- add W_ih @ x + biases, LSTM cell elementwise ----
  float g4[4];
  #pragma unroll
  for (int q = 0; q < 4; ++q) {
    const int g = q * H + t;
    float acc = b_ih[g] + b_hh[g] + lds_gates[g];
    const float* wr = W_ih + (size_t)g * IN;
    #pragma unroll
    for (int j = 0; j < IN; ++j) acc += wr[j] * lds_x[j];
    g4[q] = acc;
  }
  const float ig = sigmoidf_(g4[0]);
  const float fg = sigmoidf_(g4[1]);
  const float gg = tanhf(g4[2]);
  const float og = sigmoidf_(g4[3]);
  const float c_new = fg * cn[t] + ig * gg;
  const float h_new = og * tanhf(c_new);
  out[335 + t] = c_new;   // cn' at [335,591)
  out[79 + t] = h_new;    // hn' at [79,335)
  lds_hnew[t] = h_new;
  __syncthreads();

  // ---- heads: pitch [0,63), dur [63,79) ----
  if (t < 63) {
    float acc = pitch_b[t];
    const float* wr = pitch_W + (size_t)t * H;
    for (int j = 0; j < H; ++j) acc += wr[j] * lds_hnew[j];
    out[t] = acc;
  } else if (t >= 64 && t < 80) {
    const int d = t - 64;
    const float* wr = dur_W + (size_t)d * (10 + H);
    float acc = dur_b[d];
    #pragma unroll
    for (int j = 0; j < 10; ++j) acc += wr[j] * pitch_emb[tp * 10 + j];
    for (int j = 0; j < H; ++j) acc += wr[10 + j] * lds_hnew[j];
    out[63 + d] = acc;
  }
}

extern "C" void kernel_launch(void* const* d_in, const int* in_sizes, int n_in,
                              void* d_out, int out_size, void* d_ws, size_t ws_size,
                              hipStream_t stream) {
  const int*   sec        = (const int*)  d_in[0];
  const int*   chord      = (const int*)  d_in[1];
  const int*   true_pitch = (const int*)  d_in[2];
  const int*   note       = (const int*)  d_in[3];
  const float* hn         = (const float*)d_in[4];
  const float* cn         = (const float*)d_in[5];
  const float* E          = (const float*)d_in[6];
  const float* chord_emb  = (const float*)d_in[7];
  const float* pitch_emb  = (const float*)d_in[8];
  const float* dur_emb    = (const float*)d_in[9];
  const float* W_ih       = (const float*)d_in[10];
  const float* W_hh       = (const float*)d_in[11];
  const float* b_ih       = (const float*)d_in[12];
  const float* b_hh       = (const float*)d_in[13];
  const float* pitch_W    = (const float*)d_in[14];
  const float* pitch_b    = (const float*)d_in[15];
  const float* dur_W      = (const float*)d_in[16];
  const float* dur_b      = (const float*)d_in[17];
  const float* attn_W     = (const float*)d_in[18];
  const float* attn_b     = (const float*)d_in[19];
  float* out = (float*)d_out;

  const int L = in_sizes[6] / H;  // 131072

  float* ws = (float*)d_ws;
  float* ws_m = ws;               // [NB]
  float* ws_s = ws + NB;          // [NB]
  float* ws_acc = ws + 2 * NB;    // [NB][H]

  attn_partial_kernel<<<NB, 256, 0, stream>>>(E, hn, ws_m, ws_s, ws_acc, L);
  decoder_tail_kernel<<<1, 256, 0, stream>>>(
      ws_m, ws_s, ws_acc, sec, chord, true_pitch, note, hn, cn,
      chord_emb, pitch_emb, dur_emb, W_ih, W_hh, b_ih, b_hh,
      pitch_W, pitch_b, dur_W, dur_b, attn_W, attn_b, out);
}